// FullConection_46909632806926
// MI455X (gfx1250) — compile-verified
//
#include <hip/hip_runtime.h>

// MI455X / gfx1250, wave32. Fused kernel:
//   out = aeg(x, weight) * (x @ proj_w.T + proj_b)
// GEMM half uses V_WMMA_F32_16X16X4_F32 (full fp32 precision, matrix pipe).
// aeg half is a per-element sequential recurrence -> branchless VALU chain.

typedef __attribute__((ext_vector_type(2))) float v2f;
typedef __attribute__((ext_vector_type(8))) float v8f;

#define BATCH   1024
#define IN_F    512
#define OUT_F   512
#define KC      128   // K-chunk staged in LDS
#define LDSS    132   // LDS row stride (dwords): rows 16B-aligned, bank offset 4/row

__global__ __launch_bounds__(32)
void fullconn_fused_kernel(const float* __restrict__ x,
                           const float* __restrict__ weight,   // (1, OUT_F, IN_F)
                           const float* __restrict__ proj_w,   // (OUT_F, IN_F)
                           const float* __restrict__ proj_b,   // (OUT_F)
                           float* __restrict__ out)             // (BATCH, OUT_F)
{
    __shared__ float sx[16 * LDSS];   // x tile:      16 batch rows  x KC
    __shared__ float sw[16 * LDSS];   // weight tile: 16 out cols    x KC
    __shared__ float sp[16 * LDSS];   // proj_w tile: 16 out cols    x KC

    const int lane = threadIdx.x;       // 0..31
    const int lk   = lane & 15;         // n (out-feature) within tile
    const int hi   = lane >> 4;         // half-wave selector
    const int tile = blockIdx.x;        // 0..2047
    const int row0 = (tile >> 5) << 4;  // 64 batch-tiles
    const int col0 = (tile & 31) << 4;  // 32 out-tiles

    float acc[8];
#pragma unroll
    for (int r = 0; r < 8; ++r) acc[r] = 0.0f;
    v8f c = {};                          // WMMA accumulator (proj)

    const float bias = proj_b[col0 + lk];

    for (int kc = 0; kc < IN_F; kc += KC) {
        __syncthreads();
        // ---- cooperative staging: float4-coalesced global -> LDS ----
#pragma unroll
        for (int j = 0; j < 16; ++j) {
            const float4 xv = *(const float4*)(x      + (row0 + j) * IN_F + kc + lane * 4);
            const float4 wv = *(const float4*)(weight + (col0 + j) * IN_F + kc + lane * 4);
            const float4 pv = *(const float4*)(proj_w + (col0 + j) * IN_F + kc + lane * 4);
            *(float4*)(&sx[j * LDSS + lane * 4]) = xv;
            *(float4*)(&sw[j * LDSS + lane * 4]) = wv;
            *(float4*)(&sp[j * LDSS + lane * 4]) = pv;
        }
        __syncthreads();

        // ---- aeg recurrence: sequential in k, branchless operand select ----
        // parity p = (i + k) & 1; even: r=(r+w)*x, odd: r=(r+x)*w
        int p = (lk + kc) & 1;
        for (int k = 0; k < KC; ++k) {
            const float wv = sw[lk * LDSS + k];
#pragma unroll
            for (int r = 0; r < 8; ++r) {
                const float xv  = sx[(r + hi * 8) * LDSS + k];
                const float add = p ? xv : wv;   // v_cndmask, no divergence
                const float mul = p ? wv : xv;
                acc[r] = (acc[r] + add) * mul;
            }
            p ^= 1;
        }

        // ---- proj GEMM: V_WMMA_F32_16X16X4_F32, K stepped by 4 ----
        // A 16x4 f32 layout: lane<16 -> (M=lane, K=k..k+1), lane>=16 -> (M=lane-16, K=k+2..k+3)
        // B 4x16 mirrors it; B[k][n] = proj_w[col0+n][k]
        const int koff = hi * 2;
#pragma unroll
        for (int k4 = 0; k4 < KC; k4 += 4) {
            v2f a, b;
            a.x = sx[lk * LDSS + k4 + koff];
            a.y = sx[lk * LDSS + k4 + koff + 1];
            b.x = sp[lk * LDSS + k4 + koff];
            b.y = sp[lk * LDSS + k4 + koff + 1];
            c = __builtin_amdgcn_wmma_f32_16x16x4_f32(
                    /*neg_a=*/false, a, /*neg_b=*/false, b,
                    /*c_mod=*/(short)0, c, /*reuse_a=*/false, /*reuse_b=*/false);
        }
    }

    // ---- combine: out = aeg * (proj + bias) ----
    // c[r] holds (m = r + 8*hi, n = lk); acc[r] laid out identically.
#pragma unroll
    for (int r = 0; r < 8; ++r) {
        const int row = row0 + r + hi * 8;
        out[row * OUT_F + col0 + lk] = acc[r] * (c[r] + bias);
    }
}

extern "C" void kernel_launch(void* const* d_in, const int* in_sizes, int n_in,
                              void* d_out, int out_size, void* d_ws, size_t ws_size,
                              hipStream_t stream) {
    (void)in_sizes; (void)n_in; (void)d_ws; (void)ws_size; (void)out_size;
    const float* x      = (const float*)d_in[0];
    const float* weight = (const float*)d_in[1];
    const float* proj_w = (const float*)d_in[2];
    const float* proj_b = (const float*)d_in[3];
    float* out = (float*)d_out;

    const int tiles = (BATCH / 16) * (OUT_F / 16);  // 2048 one-wave workgroups
    fullconn_fused_kernel<<<tiles, 32, 0, stream>>>(x, weight, proj_w, proj_b, out);
}